// SnapKVAttention_82265803587994
// MI455X (gfx1250) — compile-verified
//
#include <hip/hip_runtime.h>

#define H 32
#define S 8192
#define D 128
#define W 64
#define CAP 4096
#define TOPK (CAP - W)   /* 4032 */
#define SP (S - W)       /* 8128 */
#define CHUNK 32
#define BLOCK_KEYS 1024
#define NCHUNK (BLOCK_KEYS / CHUNK)
#define NBLK (S / BLOCK_KEYS)    /* 8 */
#define QSP 132          /* padded row stride for Q tile (floats) */
#define KSP 132          /* padded row stride for K chunk (floats) */
#define LGP 36           /* padded row stride for logits (floats) */
#define KSFLOATS (CHUNK * KSP)   /* floats per K buffer */
#define NEG_HUGE (-3.402823466e+38f)
#define SCALE 0.08838834764831845f  /* 1/sqrt(128) */

typedef __attribute__((ext_vector_type(2))) float v2f;
typedef __attribute__((ext_vector_type(8))) float v8f;

// ---- CDNA5 async global->LDS helpers (ASYNCcnt-tracked DMA) ----------------
__device__ __forceinline__ unsigned int lds_off(const void* p) {
  // generic (flat) shared pointer: low 32 bits are the wave-relative LDS addr
  return (unsigned int)(unsigned long long)p;
}

__device__ __forceinline__ void async_b128(unsigned int lds_dst,
                                           unsigned long long sbase,
                                           unsigned int voff) {
  asm volatile("global_load_async_to_lds_b128 %0, %1, %2"
               :: "v"(lds_dst), "v"(voff), "s"(sbase)
               : "memory");
}

__device__ __forceinline__ void wait_async0() {
  asm volatile("s_wait_asynccnt 0x0" ::: "memory");
}
__device__ __forceinline__ void wait_async4() {
  asm volatile("s_wait_asynccnt 0x4" ::: "memory");
}

// issue one 32-key x 128-float chunk: 1024 float4s, 4 per thread (256 thr)
__device__ __forceinline__ void issue_k_chunk(unsigned int ks_dst_off,
                                              unsigned long long kbase,
                                              int key0) {
#pragma unroll
  for (int t = 0; t < 4; ++t) {
    const int i = threadIdx.x + t * 256;      // float4 index in [0,1024)
    const int r = i >> 5, c4 = i & 31;
    async_b128(ks_dst_off + (unsigned int)((r * (KSP / 4) + c4) * 16),
               kbase, (unsigned int)(key0 * D * 4 + i * 16));
  }
}

// issue Q tile (last W rows): 2048 float4s, 8 per thread
__device__ __forceinline__ void issue_q_tile(unsigned int qs_dst_off,
                                             unsigned long long qbase) {
#pragma unroll
  for (int t = 0; t < 8; ++t) {
    const int i = threadIdx.x + t * 256;      // float4 index in [0,2048)
    const int r = i >> 5, c4 = i & 31;
    async_b128(qs_dst_off + (unsigned int)((r * (QSP / 4) + c4) * 16),
               qbase, (unsigned int)(i * 16));
  }
}

// ---- 64x32 logits tile via V_WMMA_F32_16X16X4_F32 --------------------------
// 8 waves: wave w -> M-tile (w>>1)*16, N-tile (w&1)*16.  K loop over D in
// steps of 4.  A layout (ISA 7.12.2, 32-bit A 16x4): lanes 0-15 hold K=0,1 in
// VGPR0/1; lanes 16-31 hold K=2,3.  B mirrored (K split by lane half).
// C/D: VGPR r, lane l -> M = r + 8*(l>>4), N = l&15.
__device__ __forceinline__ void logits_tile(const float* __restrict__ qs,
                                            const float* __restrict__ ks,
                                            float* __restrict__ lg) {
  const int tid  = threadIdx.x;
  const int wid  = tid >> 5;
  const int lane = tid & 31;
  const int half = lane >> 4;
  const int mh   = lane & 15;
  const int m_base = (wid >> 1) << 4;
  const int n_base = (wid & 1) << 4;
  v8f c = {0.f, 0.f, 0.f, 0.f, 0.f, 0.f, 0.f, 0.f};
#pragma unroll
  for (int kk = 0; kk < D; kk += 4) {
    v2f a, b;
    a.x = qs[(m_base + mh) * QSP + kk + 2 * half];
    a.y = qs[(m_base + mh) * QSP + kk + 2 * half + 1];
    b.x = ks[(n_base + mh) * KSP + kk + 2 * half];
    b.y = ks[(n_base + mh) * KSP + kk + 2 * half + 1];
    c = __builtin_amdgcn_wmma_f32_16x16x4_f32(false, a, false, b,
                                              (short)0, c, false, false);
  }
#pragma unroll
  for (int r = 0; r < 8; ++r)
    lg[(m_base + r + 8 * half) * LGP + n_base + mh] = c[r];
}

// ---- pass 1: per-row online (max, sumexp) partials -------------------------
__global__ void snapkv_pass1(const float* __restrict__ q,
                             const float* __restrict__ kmat,
                             float* __restrict__ part_m,
                             float* __restrict__ part_z) {
  extern __shared__ float smem[];
  float* qs  = smem;                       // W*QSP
  float* ks0 = qs + W * QSP;               // 2 * CHUNK*KSP (double-buffered)
  float* lg  = ks0 + 2 * KSFLOATS;         // W*LGP
  const int h = blockIdx.x;
  const int blk = blockIdx.y;

  const unsigned long long qbase =
      (unsigned long long)(q + ((size_t)h * S + (S - W)) * D);
  const unsigned long long kbase =
      (unsigned long long)(kmat + (size_t)h * S * D);
  const unsigned int qs_off  = lds_off(qs);
  const unsigned int ks0_off = lds_off(ks0);

  issue_q_tile(qs_off, qbase);                  // 8 outstanding
  issue_k_chunk(ks0_off, kbase, blk * BLOCK_KEYS);  // +4

  float m_run = NEG_HUGE, z_run = 0.f;  // owned by threads < 64 (one per row)
  for (int c = 0; c < NCHUNK; ++c) {
    const int key0 = blk * BLOCK_KEYS + c * CHUNK;
    if (c + 1 < NCHUNK) {  // prefetch next chunk into the other buffer
      issue_k_chunk(ks0_off + (unsigned int)(((c + 1) & 1) * KSFLOATS * 4),
                    kbase, key0 + CHUNK);
      wait_async4();       // in-order completion: current buffer resident
    } else {
      wait_async0();
    }
    __syncthreads();       // everyone's DMA for buf[c&1] done; lg free
    // offset arithmetic (not pointer select) keeps addrspace(3) provable
    logits_tile(qs, ks0 + (c & 1) * KSFLOATS, lg);
    __syncthreads();
    if (threadIdx.x < W) {
      const int wrow = threadIdx.x;
      float mc = NEG_HUGE;
      for (int j = 0; j < CHUNK; ++j) {
        int kg = key0 + j;
        float v = lg[wrow * LGP + j] * SCALE;
        if (kg >= SP && (kg - SP) > wrow) v = NEG_HUGE;  // causal mask
        mc = fmaxf(mc, v);
      }
      if (mc > NEG_HUGE) {
        float zc = 0.f;
        for (int j = 0; j < CHUNK; ++j) {
          int kg = key0 + j;
          if (kg >= SP && (kg - SP) > wrow) continue;
          zc += __expf(lg[wrow * LGP + j] * SCALE - mc);
        }
        float mn = fmaxf(m_run, mc);
        z_run = z_run * __expf(m_run - mn) + zc * __expf(mc - mn);
        m_run = mn;
      }
    }
  }
  if (threadIdx.x < W) {
    const int p = (h * NBLK + blk) * W + threadIdx.x;
    part_m[p] = m_run;
    part_z[p] = z_run;
  }
}

// ---- combine block partials into final (m, 1/Z) per row --------------------
__global__ void snapkv_combine(const float* __restrict__ part_m,
                               const float* __restrict__ part_z,
                               float* __restrict__ mfin,
                               float* __restrict__ zinv) {
  const int h = blockIdx.x;
  const int w = threadIdx.x;  // 64 threads
  float m = NEG_HUGE;
  for (int b = 0; b < NBLK; ++b)
    m = fmaxf(m, part_m[(h * NBLK + b) * W + w]);
  float z = 0.f;
  for (int b = 0; b < NBLK; ++b) {
    const int p = (h * NBLK + b) * W + w;
    z += part_z[p] * __expf(part_m[p] - m);
  }
  mfin[h * W + w] = m;
  zinv[h * W + w] = 1.0f / z;
}

// ---- pass 2: recompute logits, accumulate per-key attention mass ----------
__global__ void snapkv_pass2(const float* __restrict__ q,
                             const float* __restrict__ kmat,
                             const float* __restrict__ mfin,
                             const float* __restrict__ zinv,
                             float* __restrict__ mass) {
  extern __shared__ float smem[];
  float* qs  = smem;
  float* ks0 = qs + W * QSP;
  float* lg  = ks0 + 2 * KSFLOATS;
  float* sm_m  = lg + W * LGP;   // 64
  float* sm_zi = sm_m + W;       // 64
  const int h = blockIdx.x;
  const int blk = blockIdx.y;

  const unsigned long long qbase =
      (unsigned long long)(q + ((size_t)h * S + (S - W)) * D);
  const unsigned long long kbase =
      (unsigned long long)(kmat + (size_t)h * S * D);
  const unsigned int qs_off  = lds_off(qs);
  const unsigned int ks0_off = lds_off(ks0);

  issue_q_tile(qs_off, qbase);
  issue_k_chunk(ks0_off, kbase, blk * BLOCK_KEYS);
  if (threadIdx.x < W) {
    sm_m[threadIdx.x]  = mfin[h * W + threadIdx.x];
    sm_zi[threadIdx.x] = zinv[h * W + threadIdx.x];
  }

  for (int c = 0; c < NCHUNK; ++c) {
    const int key0 = blk * BLOCK_KEYS + c * CHUNK;
    if (key0 >= SP) break;  // uniform: remaining chunks are window-only keys
    if (c + 1 < NCHUNK) {
      issue_k_chunk(ks0_off + (unsigned int)(((c + 1) & 1) * KSFLOATS * 4),
                    kbase, key0 + CHUNK);
      wait_async4();
    } else {
      wait_async0();
    }
    __syncthreads();
    logits_tile(qs, ks0 + (c & 1) * KSFLOATS, lg);
    __syncthreads();
    if (threadIdx.x < CHUNK) {
      const int j = threadIdx.x;
      const int kg = key0 + j;
      if (kg < SP) {
        float acc = 0.f;
        for (int w = 0; w < W; ++w)
          acc += __expf(lg[w * LGP + j] * SCALE - sm_m[w]) * sm_zi[w];
        mass[(size_t)h * S + kg] = acc;
      }
    }
  }
}

// ---- 13-tap average pool, same padding, count_include_pad ------------------
__global__ void snapkv_pool(const float* __restrict__ mass,
                            float* __restrict__ pooled) {
  const int i = blockIdx.x * blockDim.x + threadIdx.x;
  if (i >= H * SP) return;
  const int h = i / SP, s = i % SP;
  const float* mh = mass + (size_t)h * S;
  float acc = 0.f;
#pragma unroll
  for (int t = -6; t <= 6; ++t) {
    int p = s + t;
    if (p >= 0 && p < SP) acc += mh[p];
  }
  pooled[i] = acc * (1.0f / 13.0f);
}

// ---- full bitonic sort (value desc, index asc ties) -> top 4032 indices ----
__global__ void snapkv_topk(const float* __restrict__ pooled,
                            int* __restrict__ topidx) {
  extern __shared__ char smraw[];
  float* val = (float*)smraw;                    // 8192 floats
  int* idx = (int*)(smraw + S * sizeof(float));  // 8192 ints
  const int h = blockIdx.x;
  for (int e = threadIdx.x; e < S; e += blockDim.x) {
    if (e < SP) { val[e] = pooled[h * SP + e]; idx[e] = e; }
    else        { val[e] = NEG_HUGE;           idx[e] = 0x40000000; }
  }
  __syncthreads();
  for (int k = 2; k <= S; k <<= 1) {
    for (int j = k >> 1; j > 0; j >>= 1) {
      for (int e = threadIdx.x; e < S; e += blockDim.x) {
        const int p = e ^ j;
        if (p > e) {
          const bool desc = ((e & k) == 0);
          float v1 = val[e], v2 = val[p];
          int i1 = idx[e], i2 = idx[p];
          // "e should come first" in descending order (ties: lower idx first)
          const bool eFirst = (v1 > v2) || (v1 == v2 && i1 < i2);
          if (desc ? !eFirst : eFirst) {
            val[e] = v2; val[p] = v1;
            idx[e] = i2; idx[p] = i1;
          }
        }
      }
      __syncthreads();
    }
  }
  for (int e = threadIdx.x; e < TOPK; e += blockDim.x)
    topidx[h * TOPK + e] = idx[e];
}

// ---- gather selected K/V rows + window rows into output --------------------
__global__ void snapkv_gather(const float* __restrict__ kmat,
                              const float* __restrict__ vmat,
                              const int* __restrict__ topidx,
                              float* __restrict__ out) {
  const int row = blockIdx.x * 8 + (threadIdx.x >> 5);
  const int lane = threadIdx.x & 31;
  if (row >= H * CAP) return;
  const int h = row / CAP, j = row % CAP;
  const int src = (j < TOPK) ? topidx[h * TOPK + j] : (SP + (j - TOPK));
  const float4* ksrc = (const float4*)(kmat + ((size_t)h * S + src) * D);
  const float4* vsrc = (const float4*)(vmat + ((size_t)h * S + src) * D);
  float4* kdst = (float4*)(out + ((size_t)h * CAP + j) * D);
  float4* vdst = (float4*)(out + (size_t)H * CAP * D + ((size_t)h * CAP + j) * D);
  kdst[lane] = ksrc[lane];
  vdst[lane] = vsrc[lane];
}

extern "C" void kernel_launch(void* const* d_in, const int* in_sizes, int n_in,
                              void* d_out, int out_size, void* d_ws, size_t ws_size,
                              hipStream_t stream) {
  const float* q = (const float*)d_in[0];
  const float* k = (const float*)d_in[1];
  const float* v = (const float*)d_in[2];
  float* out = (float*)d_out;

  // workspace carve (floats)
  float* part_m = (float*)d_ws;                       // H*NBLK*W = 16384
  float* part_z = part_m + H * NBLK * W;              // 16384
  float* mfin   = part_z + H * NBLK * W;              // 2048
  float* zinv   = mfin + H * W;                       // 2048
  float* mass   = zinv + H * W;                       // H*S = 262144
  float* pooled = mass + (size_t)H * S;               // H*SP = 260096
  int*   topidx = (int*)(pooled + (size_t)H * SP);    // H*TOPK ints

  const int smem_pass =
      (W * QSP + 2 * KSFLOATS + W * LGP + 2 * W) * sizeof(float);
  const int smem_sort = S * (sizeof(float) + sizeof(int));  // 64 KB

  dim3 gpass(H, NBLK);
  snapkv_pass1<<<gpass, 256, smem_pass, stream>>>(q, k, part_m, part_z);
  snapkv_combine<<<H, W, 0, stream>>>(part_m, part_z, mfin, zinv);
  snapkv_pass2<<<gpass, 256, smem_pass, stream>>>(q, k, mfin, zinv, mass);
  snapkv_pool<<<(H * SP + 255) / 256, 256, 0, stream>>>(mass, pooled);
  snapkv_topk<<<H, 1024, smem_sort, stream>>>(pooled, topidx);
  snapkv_gather<<<(H * CAP) / 8, 256, 0, stream>>>(k, v, topidx, out);
}